// DescrptSeA_9766755631235
// MI455X (gfx1250) — compile-verified
//
#include <hip/hip_runtime.h>
#include <hip/hip_bf16.h>
#include <math.h>

typedef __attribute__((ext_vector_type(2))) float v2f;
typedef __attribute__((ext_vector_type(8))) float v8f;

#define NTYPES 2
#define SEL0   46
#define SEL1   92
#define NNEI   138
#define MP     144   // padded rows: seg0 rows [0,48) (46 valid), seg1 rows [48,144) (92 valid)
#define N1     24
#define N2     48
#define N3     96
#define AXIS   8
#define NATOMS 4096
#define NF     2
#define BLOCK  256

__device__ __forceinline__ v8f wmma_f32_k4(v2f a, v2f b, v8f c) {
  // D(16x16,f32) = A(16x4,f32) * B(4x16,f32) + C
  return __builtin_amdgcn_wmma_f32_16x16x4_f32(false, a, false, b, (short)0, c,
                                               false, false);
}

__global__ void __launch_bounds__(BLOCK)
descrpt_sea_kernel(const float* __restrict__ coord,  // (NF,NATOMS,3)
                   const float* __restrict__ davg,   // (NTYPES,NNEI,4)
                   const float* __restrict__ dstd,   // (NTYPES,NNEI,4)
                   const float* __restrict__ W1g,    // (NTYPES,1,24)
                   const float* __restrict__ b1g,    // (NTYPES,24)
                   const float* __restrict__ W2g,    // (NTYPES,24,48)
                   const float* __restrict__ b2g,    // (NTYPES,48)
                   const float* __restrict__ W3g,    // (NTYPES,48,96)
                   const float* __restrict__ b3g,    // (NTYPES,96)
                   const int*   __restrict__ atype,  // (NF,NATOMS)
                   const int*   __restrict__ nlist,  // (NF,NATOMS,NNEI)
                   float*       __restrict__ out)    // (NF,NATOMS,96*8)
{
  extern __shared__ float smem[];
  float* sW2 = smem;                    // [2][24][48]
  float* sW3 = sW2 + NTYPES * N1 * N2;  // [2][48][96]
  float* sW1 = sW3 + NTYPES * N2 * N3;  // [2][24]
  float* sb1 = sW1 + NTYPES * N1;       // [2][24]
  float* sb2 = sb1 + NTYPES * N1;       // [2][48]
  float* sb3 = sb2 + NTYPES * N2;       // [2][96]
  float* sd  = sb3 + NTYPES * N3;       // [144][4]
  float* sh1 = sd  + MP * 4;            // [144][24]
  float* sx2 = sh1 + MP * N1;           // [144][48]
  float* sG  = sx2 + MP * N2;           // [144][96]
  float* sgr = sG  + MP * N3;           // [4][96]

  const int tid = threadIdx.x;
  const int bid = blockIdx.x;
  const int f   = bid >> 12;            // NATOMS == 4096
  const int n   = bid & (NATOMS - 1);
  const int an  = f * NATOMS + n;

  // ---- phase 0: stage weights into LDS ----
  for (int i = tid; i < NTYPES * N1 * N2; i += BLOCK) sW2[i] = W2g[i];
  for (int i = tid; i < NTYPES * N2 * N3; i += BLOCK) sW3[i] = W3g[i];
  if (tid < NTYPES * N1) { sW1[tid] = W1g[tid]; sb1[tid] = b1g[tid]; }
  if (tid < NTYPES * N2) sb2[tid] = b2g[tid];
  if (tid < NTYPES * N3) sb3[tid] = b3g[tid];

  // ---- phase 1: environment matrix d (padded to MP rows) ----
  const int   ct = atype[an];
  const float cx = coord[an * 3 + 0];
  const float cy = coord[an * 3 + 1];
  const float cz = coord[an * 3 + 2];
  for (int row = tid; row < MP; row += BLOCK) {
    const bool valid = (row < SEL0) || (row >= 48 && row < 48 + SEL1);
    float e0 = 0.f, e1 = 0.f, e2 = 0.f, e3 = 0.f;
    if (valid) {
      const int nei = (row < 48) ? row : row - 2;
      const int j   = nlist[an * NNEI + nei];
      const int jn  = f * NATOMS + j;
      const float dx = coord[jn * 3 + 0] - cx;
      const float dy = coord[jn * 3 + 1] - cy;
      const float dz = coord[jn * 3 + 2] - cz;
      const float rsq = dx * dx + dy * dy + dz * dz;
      const float r   = sqrtf(fmaxf(rsq, 1e-12f));
      const float uu  = (r - 0.5f) * (1.0f / 5.5f);
      const float vv  = uu * uu * uu * (uu * (-6.0f * uu + 15.0f) - 10.0f) + 1.0f;
      const float sw  = (r < 0.5f) ? 1.0f : ((r < 6.0f) ? vv : 0.0f);
      const float s   = sw / r;
      const float sr  = s / r;
      const int   ib  = (ct * NNEI + nei) * 4;
      e0 = (s       - davg[ib + 0]) / dstd[ib + 0];
      e1 = (sr * dx - davg[ib + 1]) / dstd[ib + 1];
      e2 = (sr * dy - davg[ib + 2]) / dstd[ib + 2];
      e3 = (sr * dz - davg[ib + 3]) / dstd[ib + 3];
    }
    sd[row * 4 + 0] = e0;
    sd[row * 4 + 1] = e1;
    sd[row * 4 + 2] = e2;
    sd[row * 4 + 3] = e3;
  }
  __syncthreads();

  // ---- phase 2: h1 = tanh(s * W1[t] + b1[t]) ----
  for (int idx = tid; idx < MP * N1; idx += BLOCK) {
    const int row = idx / N1;
    const int col = idx - row * N1;
    const int t   = (row >= 48) ? 1 : 0;
    const float s = sd[row * 4];
    sh1[idx] = tanhf(s * sW1[t * N1 + col] + sb1[t * N1 + col]);
  }
  __syncthreads();

  const int wave = tid >> 5;
  const int lane = tid & 31;
  const int lo   = lane & 15;
  const int hi   = lane >> 4;

  // ---- phase 3: h2 = tanh(h1 @ W2 + b2); x2 = [h1,h1] + h2 ----
  for (int tile = wave; tile < 9 * 3; tile += 8) {
    const int mt = tile / 3;
    const int nt = tile - mt * 3;
    const int t  = (mt >= 3) ? 1 : 0;
    const float* Bw = sW2 + t * N1 * N2;
    v8f acc = {0, 0, 0, 0, 0, 0, 0, 0};
    for (int k0 = 0; k0 < N1; k0 += 4) {
      const int ka = k0 + 2 * hi;
      v2f a, b;
      a.x = sh1[(mt * 16 + lo) * N1 + ka];
      a.y = sh1[(mt * 16 + lo) * N1 + ka + 1];
      b.x = Bw[ka * N2 + nt * 16 + lo];
      b.y = Bw[(ka + 1) * N2 + nt * 16 + lo];
      acc = wmma_f32_k4(a, b, acc);
    }
    for (int v = 0; v < 8; ++v) {
      const int row = mt * 16 + v + 8 * hi;
      const int col = nt * 16 + lo;
      const int c1  = (col >= N1) ? (col - N1) : col;
      sx2[row * N2 + col] =
          tanhf(acc[v] + sb2[t * N2 + col]) + sh1[row * N1 + c1];
    }
  }
  __syncthreads();

  // ---- phase 4: h3 = tanh(x2 @ W3 + b3); G = [x2,x2] + h3 ----
  for (int tile = wave; tile < 9 * 6; tile += 8) {
    const int mt = tile / 6;
    const int nt = tile - mt * 6;
    const int t  = (mt >= 3) ? 1 : 0;
    const float* Bw = sW3 + t * N2 * N3;
    v8f acc = {0, 0, 0, 0, 0, 0, 0, 0};
    for (int k0 = 0; k0 < N2; k0 += 4) {
      const int ka = k0 + 2 * hi;
      v2f a, b;
      a.x = sx2[(mt * 16 + lo) * N2 + ka];
      a.y = sx2[(mt * 16 + lo) * N2 + ka + 1];
      b.x = Bw[ka * N3 + nt * 16 + lo];
      b.y = Bw[(ka + 1) * N3 + nt * 16 + lo];
      acc = wmma_f32_k4(a, b, acc);
    }
    for (int v = 0; v < 8; ++v) {
      const int row = mt * 16 + v + 8 * hi;
      const int col = nt * 16 + lo;
      const int c2  = (col >= N2) ? (col - N2) : col;
      sG[row * N3 + col] =
          tanhf(acc[v] + sb3[t * N3 + col]) + sx2[row * N2 + c2];
    }
  }
  __syncthreads();

  // ---- phase 5: gr(4x96) = d^T @ G / NNEI  (A = d^T: only M rows 0..3 used) ----
  if (wave < 6) {
    const int nt = wave;
    v8f acc = {0, 0, 0, 0, 0, 0, 0, 0};
    for (int k0 = 0; k0 < MP; k0 += 4) {
      const int ka = k0 + 2 * hi;
      v2f a, b;
      a.x = (lo < 4) ? sd[ka * 4 + lo] : 0.0f;
      a.y = (lo < 4) ? sd[(ka + 1) * 4 + lo] : 0.0f;
      b.x = sG[ka * N3 + nt * 16 + lo];
      b.y = sG[(ka + 1) * N3 + nt * 16 + lo];
      acc = wmma_f32_k4(a, b, acc);
    }
    if (hi == 0) {
      for (int v = 0; v < 4; ++v)
        sgr[v * N3 + nt * 16 + lo] = acc[v] * (1.0f / (float)NNEI);
    }
  }
  __syncthreads();

  // ---- phase 6: D(96x8) = gr^T @ gr[:, :AXIS] ----
  float* o = out + (size_t)an * (N3 * AXIS);
  for (int idx = tid; idx < N3 * AXIS; idx += BLOCK) {
    const int m = idx >> 3;
    const int k = idx & 7;
    float dsum = 0.f;
    for (int a2 = 0; a2 < 4; ++a2)
      dsum = fmaf(sgr[a2 * N3 + m], sgr[a2 * N3 + k], dsum);
    o[idx] = dsum;
  }
}

extern "C" void kernel_launch(void* const* d_in, const int* in_sizes, int n_in,
                              void* d_out, int out_size, void* d_ws,
                              size_t ws_size, hipStream_t stream) {
  (void)in_sizes; (void)n_in; (void)d_ws; (void)ws_size; (void)out_size;
  const float* coord = (const float*)d_in[0];
  const float* davg  = (const float*)d_in[1];
  const float* dstd  = (const float*)d_in[2];
  const float* W1    = (const float*)d_in[3];
  const float* b1    = (const float*)d_in[4];
  const float* W2    = (const float*)d_in[5];
  const float* b2    = (const float*)d_in[6];
  const float* W3    = (const float*)d_in[7];
  const float* b3    = (const float*)d_in[8];
  const int*   atype = (const int*)d_in[9];
  const int*   nlist = (const int*)d_in[10];
  float*       outp  = (float*)d_out;

  const size_t shbytes =
      (size_t)(NTYPES * N1 * N2 + NTYPES * N2 * N3 + 2 * NTYPES * N1 +
               NTYPES * N2 + NTYPES * N3 + MP * 4 + MP * N1 + MP * N2 +
               MP * N3 + 4 * N3) * sizeof(float);   // 148224 bytes

  dim3 grid(NF * NATOMS);
  dim3 block(BLOCK);
  descrpt_sea_kernel<<<grid, block, shbytes, stream>>>(
      coord, davg, dstd, W1, b1, W2, b2, W3, b3, atype, nlist, outp);
}